// TransformerBlock_28441273434139
// MI455X (gfx1250) — compile-verified
//
#include <hip/hip_runtime.h>
#include <hip/hip_bf16.h>
#include <math.h>

#define NN 100000
#define EE 1250000
#define DD 64
#define TILES (NN / 16)        // 6250, N is a multiple of 16
#define WAVES_PER_BLOCK 8

typedef float v2f __attribute__((ext_vector_type(2)));
typedef float v8f __attribute__((ext_vector_type(8)));

// Native fp32 WMMA: D(16x16) = A(16x4) * B(4x16) + C.  8-arg form:
// (neg_a, A, neg_b, B, c_mod, C, reuse_a, reuse_b)
__device__ __forceinline__ v8f wmma4_f32(v2f a, v2f b, v8f c) {
    return __builtin_amdgcn_wmma_f32_16x16x4_f32(
        false, a, false, b, (short)0, c, false, false);
}

// B fragment for row-major weight W[64][64]: k-step k (rows 4k..4k+3),
// n-tile column col = n*16 + (lane&15), lane half kh = lane>>4.
// vgpr0 holds B rows {4k+2kh}, vgpr1 holds rows {4k+2kh+1}.
__device__ __forceinline__ v2f ldB(const float* __restrict__ W, int krow, int col) {
    v2f b;
    b.x = W[(size_t)krow * DD + col];
    b.y = W[(size_t)(krow + 1) * DD + col];
    return b;
}

// Monotone float->uint key for atomicMax-based segment max.
__device__ __forceinline__ unsigned fkey(float f) {
    unsigned u = __float_as_uint(f);
    return (u & 0x80000000u) ? ~u : (u | 0x80000000u);
}
__device__ __forceinline__ float fdecode(unsigned k) {
    unsigned u = (k & 0x80000000u) ? (k & 0x7fffffffu) : ~k;
    return __uint_as_float(u);
}

// ---------------------------------------------------------------- zero scratch
__global__ void zero_kernel(unsigned* __restrict__ p, size_t n) {
    size_t i = (size_t)blockIdx.x * blockDim.x + threadIdx.x;
    size_t stride = (size_t)gridDim.x * blockDim.x;
    for (; i < n; i += stride) p[i] = 0u;
}

// ---------------------------------------------- node projections (4 GEMMs fused)
// h = relu(x@W_in + b_in);  x_lin = h@W_lin;  a_src = h@W_src;  a_dst = h@W_dst
__global__ __launch_bounds__(256, 1) void node_proj_kernel(
    const float* __restrict__ x,
    const float* __restrict__ W_in, const float* __restrict__ b_in,
    const float* __restrict__ W_lin, const float* __restrict__ W_src,
    const float* __restrict__ W_dst,
    float* __restrict__ x_lin, float* __restrict__ a_src, float* __restrict__ a_dst)
{
    __shared__ float hlds[WAVES_PER_BLOCK][16][68];  // +4 pad: conflicts & 8B align
    const int wave = threadIdx.x >> 5;
    const int lane = threadIdx.x & 31;
    const int tile = blockIdx.x * WAVES_PER_BLOCK + wave;
    if (tile >= TILES) return;                    // whole-wave exit: EXEC stays all-1s
    const int m  = lane & 15;                     // row within tile (A/B col)
    const int kh = lane >> 4;                     // K half select
    const int row0 = tile * 16;

    // ---- GEMM 1: h = x @ W_in ----
    v8f acc[4] = {v8f{0}, v8f{0}, v8f{0}, v8f{0}};
    const float* xrow = x + (size_t)(row0 + m) * DD;
#pragma unroll
    for (int k = 0; k < 16; ++k) {
        const int krow = 4 * k + 2 * kh;
        v2f a = *(const v2f*)(xrow + krow);
#pragma unroll
        for (int n = 0; n < 4; ++n)
            acc[n] = wmma4_f32(a, ldB(W_in, krow, n * 16 + m), acc[n]);
    }
    // bias + relu, stage h tile to LDS (same-wave LDS is kept in order)
#pragma unroll
    for (int n = 0; n < 4; ++n) {
        const int col = n * 16 + m;
        const float bias = b_in[col];
#pragma unroll
        for (int r = 0; r < 8; ++r)
            hlds[wave][r + 8 * kh][col] = fmaxf(acc[n][r] + bias, 0.0f);
    }

    // ---- GEMMs 2-4: x_lin / a_src / a_dst = h @ W ----
    v8f accL[4] = {v8f{0}, v8f{0}, v8f{0}, v8f{0}};
    v8f accS[4] = {v8f{0}, v8f{0}, v8f{0}, v8f{0}};
    v8f accD[4] = {v8f{0}, v8f{0}, v8f{0}, v8f{0}};
#pragma unroll
    for (int k = 0; k < 16; ++k) {
        const int krow = 4 * k + 2 * kh;
        v2f a = *(const v2f*)(&hlds[wave][m][krow]);
#pragma unroll
        for (int n = 0; n < 4; ++n) {
            const int col = n * 16 + m;
            accL[n] = wmma4_f32(a, ldB(W_lin, krow, col), accL[n]);
            accS[n] = wmma4_f32(a, ldB(W_src, krow, col), accS[n]);
            accD[n] = wmma4_f32(a, ldB(W_dst, krow, col), accD[n]);
        }
    }
#pragma unroll
    for (int n = 0; n < 4; ++n) {
        const int col = n * 16 + m;
#pragma unroll
        for (int r = 0; r < 8; ++r) {
            const size_t o = (size_t)(row0 + r + 8 * kh) * DD + col;
            x_lin[o] = accL[n][r];
            a_src[o] = accS[n][r];
            a_dst[o] = accD[n][r];
        }
    }
}

// ------------------------------------------------- edge pass 1: segment max
// 4 threads per edge, 16 channels each (64B coalesced chunks per lane group).
__global__ __launch_bounds__(256, 1) void edge_max_kernel(
    const int* __restrict__ ei, const float* __restrict__ pos,
    const float* __restrict__ W_pos, const float* __restrict__ b_pos,
    const float* __restrict__ a_src, const float* __restrict__ a_dst,
    unsigned* __restrict__ key)
{
    const int t = blockIdx.x * 256 + threadIdx.x;
    if (t >= EE * 4) return;
    const int e = t >> 2, q = t & 3;
    const int s = ei[e], d = ei[EE + e];
    const float dp0 = pos[d * 3 + 0] - pos[s * 3 + 0];
    const float dp1 = pos[d * 3 + 1] - pos[s * 3 + 1];
    const float dp2 = pos[d * 3 + 2] - pos[s * 3 + 2];
    const size_t os = (size_t)s * DD + q * 16;
    const size_t od = (size_t)d * DD + q * 16;
#pragma unroll
    for (int c = 0; c < 16; ++c) {
        const int col = q * 16 + c;
        const float delta = fmaf(dp0, W_pos[col],
                            fmaf(dp1, W_pos[DD + col],
                            fmaf(dp2, W_pos[2 * DD + col], b_pos[col])));
        const float alpha = a_dst[od + c] - a_src[os + c] + delta;
        atomicMax(key + od + c, fkey(alpha));
    }
}

// --------------------------------- edge pass 2: exp + numerator/denominator
__global__ __launch_bounds__(256, 1) void edge_acc_kernel(
    const int* __restrict__ ei, const float* __restrict__ pos,
    const float* __restrict__ W_pos, const float* __restrict__ b_pos,
    const float* __restrict__ a_src, const float* __restrict__ a_dst,
    const float* __restrict__ x_lin, const unsigned* __restrict__ key,
    float* __restrict__ ssum, float* __restrict__ snum)
{
    const int t = blockIdx.x * 256 + threadIdx.x;
    if (t >= EE * 4) return;
    const int e = t >> 2, q = t & 3;
    const int s = ei[e], d = ei[EE + e];
    const float dp0 = pos[d * 3 + 0] - pos[s * 3 + 0];
    const float dp1 = pos[d * 3 + 1] - pos[s * 3 + 1];
    const float dp2 = pos[d * 3 + 2] - pos[s * 3 + 2];
    const size_t os = (size_t)s * DD + q * 16;
    const size_t od = (size_t)d * DD + q * 16;
#pragma unroll
    for (int c = 0; c < 16; ++c) {
        const int col = q * 16 + c;
        const float delta = fmaf(dp0, W_pos[col],
                            fmaf(dp1, W_pos[DD + col],
                            fmaf(dp2, W_pos[2 * DD + col], b_pos[col])));
        const float alpha = a_dst[od + c] - a_src[os + c] + delta;
        const float ex = __expf(alpha - fdecode(key[od + c]));
        atomicAdd(ssum + od + c, ex);
        atomicAdd(snum + od + c, ex * (x_lin[os + c] + delta));
    }
}

// --------------------------------- output: relu((num/(den+eps)) @ W_out + b_out)
__global__ __launch_bounds__(256, 1) void node_out_kernel(
    const float* __restrict__ snum, const float* __restrict__ ssum,
    const float* __restrict__ W_out, const float* __restrict__ b_out,
    float* __restrict__ out)
{
    const int wave = threadIdx.x >> 5;
    const int lane = threadIdx.x & 31;
    const int tile = blockIdx.x * WAVES_PER_BLOCK + wave;
    if (tile >= TILES) return;
    const int m  = lane & 15;
    const int kh = lane >> 4;
    const int row0 = tile * 16;

    const float* nrow = snum + (size_t)(row0 + m) * DD;
    const float* drow = ssum + (size_t)(row0 + m) * DD;
    v8f acc[4] = {v8f{0}, v8f{0}, v8f{0}, v8f{0}};
#pragma unroll
    for (int k = 0; k < 16; ++k) {
        const int krow = 4 * k + 2 * kh;
        v2f nv = *(const v2f*)(nrow + krow);
        v2f dv = *(const v2f*)(drow + krow);
        v2f a;
        a.x = nv.x / (dv.x + 1e-16f);
        a.y = nv.y / (dv.y + 1e-16f);
#pragma unroll
        for (int n = 0; n < 4; ++n)
            acc[n] = wmma4_f32(a, ldB(W_out, krow, n * 16 + m), acc[n]);
    }
#pragma unroll
    for (int n = 0; n < 4; ++n) {
        const int col = n * 16 + m;
        const float bias = b_out[col];
#pragma unroll
        for (int r = 0; r < 8; ++r)
            out[(size_t)(row0 + r + 8 * kh) * DD + col] = fmaxf(acc[n][r] + bias, 0.0f);
    }
}

extern "C" void kernel_launch(void* const* d_in, const int* in_sizes, int n_in,
                              void* d_out, int out_size, void* d_ws, size_t ws_size,
                              hipStream_t stream)
{
    const float* x     = (const float*)d_in[0];
    const float* pos   = (const float*)d_in[1];
    const int*   ei    = (const int*)d_in[2];   // [2,E]: src row then dst row
    const float* W_in  = (const float*)d_in[3];
    const float* b_in  = (const float*)d_in[4];
    const float* W_lin = (const float*)d_in[5];
    const float* W_src = (const float*)d_in[6];
    const float* W_dst = (const float*)d_in[7];
    const float* W_pos = (const float*)d_in[8];
    const float* b_pos = (const float*)d_in[9];
    const float* W_out = (const float*)d_in[10];
    const float* b_out = (const float*)d_in[11];

    const size_t nd = (size_t)NN * DD;
    float*    x_lin = (float*)d_ws;
    float*    a_src = x_lin + nd;
    float*    a_dst = a_src + nd;
    unsigned* key   = (unsigned*)(a_dst + nd);
    float*    ssum  = (float*)(key + nd);
    float*    snum  = ssum + nd;

    // zero the three accumulator arrays (key, ssum, snum are contiguous)
    zero_kernel<<<4096, 256, 0, stream>>>(key, 3 * nd);

    const int node_blocks = (TILES + WAVES_PER_BLOCK - 1) / WAVES_PER_BLOCK;  // 782
    node_proj_kernel<<<node_blocks, 256, 0, stream>>>(
        x, W_in, b_in, W_lin, W_src, W_dst, x_lin, a_src, a_dst);

    const int edge_blocks = (EE * 4 + 255) / 256;
    edge_max_kernel<<<edge_blocks, 256, 0, stream>>>(
        ei, pos, W_pos, b_pos, a_src, a_dst, key);
    edge_acc_kernel<<<edge_blocks, 256, 0, stream>>>(
        ei, pos, W_pos, b_pos, a_src, a_dst, x_lin, key, ssum, snum);

    node_out_kernel<<<node_blocks, 256, 0, stream>>>(
        snum, ssum, W_out, b_out, (float*)d_out);
}